// LocalNeighborhood_2456721293910
// MI455X (gfx1250) — compile-verified
//
#include <hip/hip_runtime.h>

typedef __attribute__((ext_vector_type(2))) float v2f;
typedef __attribute__((ext_vector_type(8))) float v8f;

#define NPTS 4096
#define NB   8
#define KMAX 16

// frame element (b,n,r,d) lives at ((b*NPTS + n)*4 + r)*3 + d   (12 floats per point)
// out = [euclid (8,4096,16,3)] ++ [dist (8,4096,16,1)] ++ [nei_attr (8,4096,16,128)]

__global__ __launch_bounds__(256)
void knn_wmma_kernel(const float* __restrict__ frame,
                     const float* __restrict__ attr,
                     float* __restrict__ out)
{
    const int lane = threadIdx.x & 31;
    const int wave = threadIdx.x >> 5;
    const int tile = blockIdx.x * 8 + wave;     // 0..2047
    const int b    = tile >> 8;                 // batch
    const int q0   = (tile & 255) * 16;         // first query of this tile

    __shared__ float ldsD[8][256];              // 16x16 distance tile per wave
    __shared__ int   ldsIdx[8][256];            // 16 queries x 16 neighbor indices per wave
    float* Dtile  = ldsD[wave];
    int*   IdxRow = ldsIdx[wave];

    const int half = lane >> 4;                 // which 16-lane half
    const int l16  = lane & 15;

    const float* fb = frame + (size_t)b * NPTS * 12;

    // ---- A operand: 16x4 query tile (K=3 padded to 4). Layout (ISA 7.12.2):
    // lane l: M = l%16 ; VGPR j: K = 2*(l>=16) + j
    const int qm = q0 + l16;
    const float* qc = fb + (size_t)qm * 12;     // center of query M=l16
    const int k1 = 2 * half, k2 = 2 * half + 1;
    v2f A;
    A.x = qc[k1];                               // k1 in {0,2} -> always valid
    A.y = (k2 < 3) ? qc[k2] : 0.0f;             // k2 == 3 -> zero pad

    // Query norms for the C-tile rows this half-wave owns: M = r + 8*half
    float nq[8];
#pragma unroll
    for (int r = 0; r < 8; ++r) {
        const float* c = fb + (size_t)(q0 + 8 * half + r) * 12;
        nq[r] = c[0] * c[0] + c[1] * c[1] + c[2] * c[2];
    }

    // ---- top-K state (lanes 0..15 use it; one query row per lane)
    float bd[KMAX];
    int   bi[KMAX];
#pragma unroll
    for (int k = 0; k < KMAX; ++k) { bd[k] = 3.0e38f; bi[k] = 0; }
    float worst = 3.0e38f;
    int   wslot = 0;

    const v8f czero = {};

    // ---- stream all candidate tiles
    for (int ct = 0; ct < NPTS / 16; ++ct) {
        const int c0 = ct * 16;
        // B operand: 4x16 candidate tile, same lane indexing pattern as A
        const float* cc = fb + (size_t)(c0 + l16) * 12;
        v2f Bm;
        Bm.x = cc[k1];
        Bm.y = (k2 < 3) ? cc[k2] : 0.0f;
        // candidate norm: C-layout column N = l16 is exactly this lane's candidate
        const float nc = cc[0] * cc[0] + cc[1] * cc[1] + cc[2] * cc[2];

        // G = A x B  (16x16 dot products, K padded to 4)
        v8f g = __builtin_amdgcn_wmma_f32_16x16x4_f32(
            false, A, false, Bm, (short)0, czero, false, false);

        // D = |q|^2 + |c|^2 - 2 q.c  -> LDS row-major 16x16
#pragma unroll
        for (int r = 0; r < 8; ++r) {
            const int M = r + 8 * half;
            Dtile[M * 16 + l16] = nq[r] + nc - 2.0f * g[r];
        }
        __builtin_amdgcn_wave_barrier();
        asm volatile("s_wait_dscnt 0" ::: "memory");

        // per-row top-16 maintenance (lanes 0..15, one query each)
        if (lane < 16) {
#pragma unroll 4
            for (int n = 0; n < 16; ++n) {
                const int cand = c0 + n;
                float d = Dtile[lane * 16 + n];
                if (cand == qm) d = 0.0f;       // exact self-distance
                if (d < worst) {
                    bd[wslot] = d; bi[wslot] = cand;
                    worst = bd[0]; wslot = 0;
#pragma unroll
                    for (int k = 1; k < KMAX; ++k)
                        if (bd[k] > worst) { worst = bd[k]; wslot = k; }
                }
            }
        }
        __builtin_amdgcn_wave_barrier();
    }

    // ---- sort + per-query outputs (lanes 0..15)
    if (lane < 16) {
        // insertion sort by (distance, index) to mimic stable argsort
        for (int i = 1; i < KMAX; ++i) {
            const float d = bd[i]; const int ix = bi[i];
            int j = i - 1;
            while (j >= 0 && (bd[j] > d || (bd[j] == d && bi[j] > ix))) {
                bd[j + 1] = bd[j]; bi[j + 1] = bi[j]; --j;
            }
            bd[j + 1] = d; bi[j + 1] = ix;
        }

        const float* ax = fb + (size_t)qm * 12 + 3;   // 3 axes, 9 floats
        const float a00 = ax[0], a01 = ax[1], a02 = ax[2];
        const float a10 = ax[3], a11 = ax[4], a12 = ax[5];
        const float a20 = ax[6], a21 = ax[7], a22 = ax[8];
        const float qx = qc[0], qy = qc[1], qz = qc[2];

        const size_t qi   = (size_t)b * NPTS + qm;
        float* Eout = out + qi * KMAX * 3;
        float* Dout = out + (size_t)NB * NPTS * KMAX * 3 + qi * KMAX;

#pragma unroll
        for (int k = 0; k < KMAX; ++k) {
            const float* nc3 = fb + (size_t)bi[k] * 12;
            const float dx = nc3[0] - qx, dy = nc3[1] - qy, dz = nc3[2] - qz;
            Eout[k * 3 + 0] = dx * a00 + dy * a01 + dz * a02;
            Eout[k * 3 + 1] = dx * a10 + dy * a11 + dz * a12;
            Eout[k * 3 + 2] = dx * a20 + dy * a21 + dz * a22;
            Dout[k] = bd[k];
            IdxRow[lane * 16 + k] = bi[k];
        }
    }
    __builtin_amdgcn_wave_barrier();
    asm volatile("s_wait_dscnt 0" ::: "memory");

    // ---- cooperative attribute gather: all 32 lanes, float4 per lane (512B per pair)
    const float* ab = attr + (size_t)b * NPTS * 128;
    float* Aout = out + (size_t)NB * NPTS * KMAX * 3
                      + (size_t)NB * NPTS * KMAX
                      + ((size_t)b * NPTS + q0) * KMAX * 128;
#pragma unroll 4
    for (int p = 0; p < 256; ++p) {             // p = qrow*16 + k
        const int idx = IdxRow[p];
        const float4 v = ((const float4*)(ab + (size_t)idx * 128))[lane];
        ((float4*)(Aout + (size_t)p * 128))[lane] = v;
    }
}

extern "C" void kernel_launch(void* const* d_in, const int* in_sizes, int n_in,
                              void* d_out, int out_size, void* d_ws, size_t ws_size,
                              hipStream_t stream)
{
    const float* frame = (const float*)d_in[0];
    const float* attr  = (const float*)d_in[1];
    float* out = (float*)d_out;
    // 2048 query tiles total, 8 waves (tiles) per 256-thread block
    knn_wmma_kernel<<<256, 256, 0, stream>>>(frame, attr, out);
}